// SparseEmbeddingHead_39659728011418
// MI455X (gfx1250) — compile-verified
//
#include <hip/hip_runtime.h>

typedef __attribute__((ext_vector_type(2))) float v2f;
typedef __attribute__((ext_vector_type(4))) float v4f;
typedef __attribute__((ext_vector_type(8))) float v8f;

#define B_   32
#define S_   512
#define H_   1024
#define V_   250002

#define KC            64              // K-chunk staged in LDS per iteration
#define PITCH         68              // padded LDS row pitch (floats): 16B-aligned rows, 4-bank stride
#define WAVES         4
#define TOK_PER_WAVE  16
#define TOK_PER_WG    (WAVES * TOK_PER_WAVE)   // 64
#define THREADS       (WAVES * 32)             // 128

// ---------------------------------------------------------------------------
// Kernel 1: zero the [B,V] output (poisoned by harness). Vectorized stores.
// ---------------------------------------------------------------------------
__global__ void zero_out_kernel(float4* __restrict__ out4, int n4) {
    int p = blockIdx.x * blockDim.x + threadIdx.x;
    if (p < n4) out4[p] = make_float4(0.f, 0.f, 0.f, 0.f);
}

// ---------------------------------------------------------------------------
// Kernel 2: GEMV via V_WMMA_F32_16X16X4_F32 (16 tokens per wave) + scatter-add.
//   token_weights[t] = relu((dot(hidden[t,:], W) + b) * mask[t])
//   out[b, ids[t]] += token_weights[t]   (atomic f32)
// ---------------------------------------------------------------------------
__global__ __launch_bounds__(THREADS)
void sparse_head_kernel(const float* __restrict__ hidden,   // [B*S, H]
                        const int*   __restrict__ ids,      // [B*S]
                        const float* __restrict__ mask,     // [B*S]
                        const float* __restrict__ Wv,       // [H]
                        const float* __restrict__ bias,     // [1]
                        float*       __restrict__ out) {    // [B, V]
    __shared__ float sW[H_];                                // 4 KB
    __shared__ float sT[WAVES * TOK_PER_WAVE * PITCH];      // 4*16*68*4 = 17 KB

    const int tid  = threadIdx.x;
    const int lane = tid & 31;
    const int wave = tid >> 5;

    // Cooperative load of W into LDS (coalesced, broadcast-read later).
    for (int i = tid; i < H_; i += THREADS) sW[i] = Wv[i];
    __syncthreads();

    const int t0 = blockIdx.x * TOK_PER_WG + wave * TOK_PER_WAVE;
    float* tile  = &sT[wave * TOK_PER_WAVE * PITCH];

    // A-fragment addressing for 32-bit 16x4 A matrix (ISA 7.12.2):
    //   lanes 0-15: M=lane,    K = {0,1} in the two VGPRs
    //   lanes16-31: M=lane-16, K = {2,3}
    const int arow  = lane & 15;
    const int akoff = (lane >> 4) * 2;

    v8f acc = {};   // 16x16 f32 C/D accumulator (dot replicated across N)

    for (int c = 0; c < H_ / KC; ++c) {
        // ---- Stage 16 x KC f32 tile into LDS: coalesced float4 loads ----
        #pragma unroll
        for (int i = 0; i < (TOK_PER_WAVE * KC / 4) / 32; ++i) {   // 8 iters
            int p  = i * 32 + lane;          // 0..255 float4 slots
            int r  = p >> 4;                 // KC/4 = 16 float4 per row
            int c4 = p & 15;
            v4f v = *(const v4f*)&hidden[(size_t)(t0 + r) * H_ + c * KC + c4 * 4];
            *(v4f*)&tile[r * PITCH + c4 * 4] = v;
        }
        __syncthreads();   // tile visible (wave-private, but keep waves in lockstep)

        // ---- 16 WMMAs over this K-chunk: D = A(16x4) * B(4x16) + C ----
        // B[k][n] = W[k] broadcast over all n, so every column of D holds the dots.
        #pragma unroll
        for (int kk = 0; kk < KC; kk += 4) {
            v2f a = *(const v2f*)&tile[arow * PITCH + kk + akoff];
            v2f b = *(const v2f*)&sW[c * KC + kk + akoff];
            acc = __builtin_amdgcn_wmma_f32_16x16x4_f32(
                      /*neg_a=*/false, a, /*neg_b=*/false, b,
                      /*c_mod=*/(short)0, acc,
                      /*reuse_a=*/false, /*reuse_b=*/false);
        }
        __syncthreads();   // done reading tile before next chunk overwrites it
    }

    // ---- Extract dots and scatter ----
    // C/D layout: lanes 0-15 VGPR j = D[M=j][N=lane]; lanes 16-31 VGPR j = D[M=8+j][N=lane-16].
    // Since D is N-replicated: lane j (j<8) takes token t0+j from acc[j];
    // lane 16+j takes token t0+8+j from acc[j].
    const int j = lane & 15;
    if (j < 8) {
        const int m = (lane >> 4) * 8 + j;
        float d = 0.f;
        #pragma unroll
        for (int q = 0; q < 8; ++q) d = (q == j) ? acc[q] : d;

        const int t  = t0 + m;
        float w = (d + bias[0]) * mask[t];
        w = fmaxf(w, 0.f);
        const int id = ids[t];
        const int bi = t >> 9;               // t / S_ (S_=512)
        atomicAdd(&out[(size_t)bi * V_ + id], w);
    }
}

// ---------------------------------------------------------------------------
extern "C" void kernel_launch(void* const* d_in, const int* in_sizes, int n_in,
                              void* d_out, int out_size, void* d_ws, size_t ws_size,
                              hipStream_t stream) {
    const float* hidden = (const float*)d_in[0];   // [B,S,H] f32
    const int*   ids    = (const int*)  d_in[1];   // [B,S] i32
    const float* mask   = (const float*)d_in[2];   // [B,S] f32
    const float* Wv     = (const float*)d_in[3];   // [1,H] f32
    const float* bias   = (const float*)d_in[4];   // [1] f32
    float*       out    = (float*)d_out;           // [B,V] f32

    const int n4 = (B_ * V_) / 4;                  // 2,000,016 float4s
    zero_out_kernel<<<(n4 + 255) / 256, 256, 0, stream>>>((float4*)out, n4);

    const int blocks = (B_ * S_) / TOK_PER_WG;     // 256
    sparse_head_kernel<<<blocks, THREADS, 0, stream>>>(hidden, ids, mask, Wv, bias, out);
}